// ROINetwork_40836549050459
// MI455X (gfx1250) — compile-verified
//
#include <hip/hip_runtime.h>
#include <stdint.h>

typedef __attribute__((ext_vector_type(2))) float v2f;
typedef __attribute__((ext_vector_type(8))) float v8f;
typedef __attribute__((__vector_size__(4 * sizeof(int)))) int vsi4;

#define NCLS    21
#define PHH     7
#define PWW     7
#define RSCALE  0.0625f
#define CONF_TH 0.05f
#define NMS_TH  0.5f
#define POSTK   100
#define PREM    2048
#define DCLAMP  4.13516655578f   // log(1000*0.0625)

#define BATCH 2
#define CHN   256
#define FH    38
#define FW    38
#define NROI  512
#define HID   1024
#define FIN   (CHN*PHH*PWW)      // 12544
#define BNR   (BATCH*NROI)       // 1024
#define NBOX  (NROI*NCLS)        // 10752
#define NSORT 16384

// ---------------- async global->LDS (gfx1250), guarded so compile never breaks
#if __has_builtin(__builtin_amdgcn_global_load_async_to_lds_b128)
#define HAVE_ASYNC_LDS 1
__device__ __forceinline__ void async_ld16(void* lds, const void* g) {
  __builtin_amdgcn_global_load_async_to_lds_b128((vsi4*)g, (vsi4*)lds, 0, 0);
}
__device__ __forceinline__ void wait_async0() {
#if __has_builtin(__builtin_amdgcn_s_wait_asynccnt)
  __builtin_amdgcn_s_wait_asynccnt(0);
#else
  asm volatile("s_wait_asynccnt 0" ::: "memory");
#endif
}
#else
#define HAVE_ASYNC_LDS 0
#endif

// ---------------------------------------------------------------- ROI pool
__global__ __launch_bounds__(256)
void roi_pool_kernel(const float* __restrict__ feat, const float* __restrict__ rois,
                     float* __restrict__ X) {
  int blk = blockIdx.x;                 // b*NROI + n
  int b = blk / NROI;
  const float* roi = rois + (size_t)blk * 4;
  float x1 = rintf(roi[0] * RSCALE), y1 = rintf(roi[1] * RSCALE);
  float x2 = rintf(roi[2] * RSCALE), y2 = rintf(roi[3] * RSCALE);
  float rw = fmaxf(x2 - x1 + 1.f, 1.f);
  float rh = fmaxf(y2 - y1 + 1.f, 1.f);
  float bh = rh / (float)PHH, bw = rw / (float)PWW;
  int hs[PHH], he[PHH], ws[PWW], we[PWW];
  #pragma unroll
  for (int p = 0; p < PHH; ++p) {
    hs[p] = (int)fminf(fmaxf(floorf((float)p * bh) + y1, 0.f), (float)FH);
    he[p] = (int)fminf(fmaxf(ceilf((float)(p + 1) * bh) + y1, 0.f), (float)FH);
    ws[p] = (int)fminf(fmaxf(floorf((float)p * bw) + x1, 0.f), (float)FW);
    we[p] = (int)fminf(fmaxf(ceilf((float)(p + 1) * bw) + x1, 0.f), (float)FW);
  }
  int c = threadIdx.x;                  // CHN == 256
  const float* fp = feat + ((size_t)b * CHN + c) * FH * FW;
  float* xr = X + (size_t)blk * FIN + (size_t)c * (PHH * PWW);
  for (int p = 0; p < PHH; ++p) {
    for (int q = 0; q < PWW; ++q) {
      bool empty = (hs[p] >= he[p]) || (ws[q] >= we[q]);
      float m = -1e30f;
      for (int h = hs[p]; h < he[p]; ++h)
        for (int w = ws[q]; w < we[q]; ++w)
          m = fmaxf(m, fp[h * FW + w]);
      xr[p * PWW + q] = empty ? 0.f : m;
    }
  }
}

// ------------------------------------------------- WMMA f32 GEMM + bias + ReLU
// Block: 256 thr (8 waves), tile 64x64, K-tile 32, double-buffered LDS.
// Wave w: rows 32*(w>>2), cols 16*(w&3); per wave two 16x16 accumulators.
// Tile staging uses GLOBAL_LOAD_ASYNC_TO_LDS_B128 when available.
#define BMT 64
#define BNT 64
#define BKT 32
#define ASTR 36
#define BSTR 68

__global__ __launch_bounds__(256)
void gemm_bias_relu(const float* __restrict__ A, const float* __restrict__ Bm,
                    const float* __restrict__ bias, float* __restrict__ Cm,
                    int M, int N, int K, int do_relu) {
  __shared__ float As[2][BMT * ASTR];
  __shared__ float Bs[2][BKT * BSTR];
  int tid  = threadIdx.x;
  int lane = tid & 31;
  int wave = tid >> 5;
  int half = lane >> 4, l16 = lane & 15;
  int wm = wave >> 2, wn = wave & 3;
  int rowBase = blockIdx.y * BMT;
  int colBase = blockIdx.x * BNT;

  v8f acc0 = {};
  v8f acc1 = {};

  int ar = tid >> 3;            // 0..31 (+32)
  int ac = (tid & 7) * 4;
  int br = tid >> 4;            // 0..15 (+16)
  int bc = (tid & 15) * 4;

  const float* apBase = A + (size_t)(rowBase + ar) * K + ac;
  const float* bpBase = Bm + (size_t)br * N + (colBase + bc);

  int nTiles = K / BKT;

  // prologue: stage tile 0 into buffer 0
#if HAVE_ASYNC_LDS
  async_ld16(&As[0][ar * ASTR + ac],        apBase);
  async_ld16(&As[0][(ar + 32) * ASTR + ac], apBase + (size_t)32 * K);
  async_ld16(&Bs[0][br * BSTR + bc],        bpBase);
  async_ld16(&Bs[0][(br + 16) * BSTR + bc], bpBase + (size_t)16 * N);
  wait_async0();
#else
  {
    float4 va0 = *(const float4*)apBase;
    float4 va1 = *(const float4*)(apBase + (size_t)32 * K);
    float4 vb0 = *(const float4*)bpBase;
    float4 vb1 = *(const float4*)(bpBase + (size_t)16 * N);
    *(float4*)&As[0][ar * ASTR + ac]        = va0;
    *(float4*)&As[0][(ar + 32) * ASTR + ac] = va1;
    *(float4*)&Bs[0][br * BSTR + bc]        = vb0;
    *(float4*)&Bs[0][(br + 16) * BSTR + bc] = vb1;
  }
#endif
  __syncthreads();

  int aRow = 32 * wm + l16;
  int colL = 16 * wn + l16;

  for (int t = 0; t < nTiles; ++t) {
    int cur = t & 1;
    bool haveNext = (t + 1) < nTiles;

#if HAVE_ASYNC_LDS
    // kick off DMA of tile t+1 into the other buffer; overlaps WMMA burst
    if (haveNext) {
      int nxt = cur ^ 1;
      int k0n = (t + 1) * BKT;
      const float* ap = apBase + k0n;
      const float* bp = bpBase + (size_t)k0n * N;
      async_ld16(&As[nxt][ar * ASTR + ac],        ap);
      async_ld16(&As[nxt][(ar + 32) * ASTR + ac], ap + (size_t)32 * K);
      async_ld16(&Bs[nxt][br * BSTR + bc],        bp);
      async_ld16(&Bs[nxt][(br + 16) * BSTR + bc], bp + (size_t)16 * N);
      if (t + 2 < nTiles) {
        int k0p = (t + 2) * BKT;
        __builtin_prefetch(apBase + k0p, 0, 0);
        __builtin_prefetch(bpBase + (size_t)k0p * N, 0, 0);
      }
    }
#else
    float4 va0, va1, vb0, vb1;
    if (haveNext) {
      int k0n = (t + 1) * BKT;
      const float* ap = apBase + k0n;
      va0 = *(const float4*)ap;
      va1 = *(const float4*)(ap + (size_t)32 * K);
      const float* bp = bpBase + (size_t)k0n * N;
      vb0 = *(const float4*)bp;
      vb1 = *(const float4*)(bp + (size_t)16 * N);
      if (t + 2 < nTiles) {
        int k0p = (t + 2) * BKT;
        __builtin_prefetch(apBase + k0p, 0, 0);
        __builtin_prefetch(bpBase + (size_t)k0p * N, 0, 0);
      }
    }
#endif

    // register-stage all fragments for this K-tile, then WMMA burst
    v2f fa0[8], fa1[8], fb[8];
    #pragma unroll
    for (int s = 0; s < 8; ++s) {
      int ka = s * 4 + 2 * half;
      fa0[s] = *(const v2f*)&As[cur][aRow * ASTR + ka];
      fa1[s] = *(const v2f*)&As[cur][(aRow + 16) * ASTR + ka];
      fb[s].x = Bs[cur][ka * BSTR + colL];
      fb[s].y = Bs[cur][(ka + 1) * BSTR + colL];
    }
    #pragma unroll
    for (int s = 0; s < 8; ++s) {
      acc0 = __builtin_amdgcn_wmma_f32_16x16x4_f32(false, fa0[s], false, fb[s],
                                                   (short)0, acc0, false, false);
      acc1 = __builtin_amdgcn_wmma_f32_16x16x4_f32(false, fa1[s], false, fb[s],
                                                   (short)0, acc1, false, false);
    }

#if HAVE_ASYNC_LDS
    if (haveNext) wait_async0();
#else
    if (haveNext) {
      int nxt = cur ^ 1;
      *(float4*)&As[nxt][ar * ASTR + ac]        = va0;
      *(float4*)&As[nxt][(ar + 32) * ASTR + ac] = va1;
      *(float4*)&Bs[nxt][br * BSTR + bc]        = vb0;
      *(float4*)&Bs[nxt][(br + 16) * BSTR + bc] = vb1;
    }
#endif
    __syncthreads();
  }

  int col = colBase + 16 * wn + l16;
  float bv = bias[col];
  int r0 = rowBase + 32 * wm + 8 * half;     // C VGPR r -> row r + 8*half
  #pragma unroll
  for (int r = 0; r < 8; ++r) {
    float v = acc0[r] + bv;
    if (do_relu) v = fmaxf(v, 0.f);
    Cm[(size_t)(r0 + r) * N + col] = v;
  }
  #pragma unroll
  for (int r = 0; r < 8; ++r) {
    float v = acc1[r] + bv;
    if (do_relu) v = fmaxf(v, 0.f);
    Cm[(size_t)(r0 + 16 + r) * N + col] = v;
  }
}

// ---------------------------------------------------------- heads + softmax
__global__ __launch_bounds__(128)
void heads_kernel(const float* __restrict__ H2,
                  const float* __restrict__ Wloc, const float* __restrict__ bloc,
                  const float* __restrict__ Wcls, const float* __restrict__ bcls,
                  float* __restrict__ locs, float* __restrict__ probs) {
  __shared__ float xrow[HID];
  __shared__ float logits[NCLS];
  int row = blockIdx.x;
  for (int i = threadIdx.x; i < HID; i += blockDim.x)
    xrow[i] = H2[(size_t)row * HID + i];
  __syncthreads();
  int t = threadIdx.x;
  if (t < NCLS * 4) {
    float s = bloc[t];
    for (int k = 0; k < HID; ++k) s += xrow[k] * Wloc[(size_t)k * (NCLS * 4) + t];
    locs[(size_t)row * (NCLS * 4) + t] = s;
  } else if (t < NCLS * 4 + NCLS) {
    int c = t - NCLS * 4;
    float s = bcls[c];
    for (int k = 0; k < HID; ++k) s += xrow[k] * Wcls[(size_t)k * NCLS + c];
    logits[c] = s;
  }
  __syncthreads();
  if (t == 0) {
    float mx = logits[0];
    for (int c = 1; c < NCLS; ++c) mx = fmaxf(mx, logits[c]);
    float sum = 0.f;
    for (int c = 0; c < NCLS; ++c) sum += expf(logits[c] - mx);
    float inv = 1.f / sum;
    for (int c = 0; c < NCLS; ++c)
      probs[(size_t)row * NCLS + c] = expf(logits[c] - mx) * inv;
  }
}

// ------------------------------------------------- decode + clip + valid mask
__global__ __launch_bounds__(256)
void decode_kernel(const float* __restrict__ rois, const float* __restrict__ locs,
                   const float* __restrict__ probs, const int* __restrict__ img_sizes,
                   float* __restrict__ boxes, float* __restrict__ mscore) {
  int gid = blockIdx.x * blockDim.x + threadIdx.x;
  if (gid >= BATCH * NBOX) return;
  int b = gid / NBOX, i = gid % NBOX;
  int n = i / NCLS, cls = i % NCLS;
  int row = b * NROI + n;
  const float* roi = rois + (size_t)row * 4;
  float cx = (roi[0] + roi[2]) * 0.5f, cy = (roi[1] + roi[3]) * 0.5f;
  float bw = roi[2] - roi[0], bh = roi[3] - roi[1];
  const float* d = locs + (size_t)row * (NCLS * 4) + cls * 4;
  float px = cx + d[0] * bw, py = cy + d[1] * bh;
  float pw = expf(fminf(d[2], DCLAMP)) * bw;
  float ph = expf(fminf(d[3], DCLAMP)) * bh;
  float x1 = px - 0.5f * pw, y1 = py - 0.5f * ph;
  float x2 = px + 0.5f * pw, y2 = py + 0.5f * ph;
  float iw = (float)img_sizes[b * 2 + 1], ih = (float)img_sizes[b * 2 + 0];
  x1 = fminf(fmaxf(x1, 0.f), iw);  x2 = fminf(fmaxf(x2, 0.f), iw);
  y1 = fminf(fmaxf(y1, 0.f), ih);  y2 = fminf(fmaxf(y2, 0.f), ih);
  float s = probs[(size_t)row * NCLS + cls];
  bool valid = (cls > 0) && ((x2 - x1) >= 1e-3f) && ((y2 - y1) >= 1e-3f) && (s > CONF_TH);
  float* bo = boxes + ((size_t)b * NBOX + i) * 4;
  bo[0] = x1; bo[1] = y1; bo[2] = x2; bo[3] = y2;
  mscore[(size_t)b * NBOX + i] = valid ? s : -1.f;
}

// -------------------------------------------------------- sort + NMS + output
__device__ __forceinline__ unsigned ford(float f) {
  unsigned u = __float_as_uint(f);
  return (u & 0x80000000u) ? ~u : (u | 0x80000000u);
}
__device__ __forceinline__ float funord(unsigned u) {
  return __uint_as_float((u & 0x80000000u) ? (u & 0x7FFFFFFFu) : ~u);
}

// Descending bitonic sort over global keys; one block, __syncthreads gives
// block-scope global visibility.
__device__ void bitonic_sort_desc(unsigned long long* key, int n, int tid, int nth) {
  for (int k = 2; k <= n; k <<= 1) {
    for (int j = k >> 1; j > 0; j >>= 1) {
      __syncthreads();
      for (int i = tid; i < n; i += nth) {
        int ixj = i ^ j;
        if (ixj > i) {
          unsigned long long a = key[i], b = key[ixj];
          bool desc = ((i & k) == 0);
          if (desc ? (a < b) : (a > b)) { key[i] = b; key[ixj] = a; }
        }
      }
    }
  }
  __syncthreads();
}

__global__ __launch_bounds__(1024)
void nms_kernel(const float* __restrict__ boxesAll, const float* __restrict__ mscoreAll,
                const int* __restrict__ orig_sizes, unsigned long long* __restrict__ keysAll,
                float* __restrict__ out) {
  __shared__ float bx[PREM * 4];
  __shared__ float sc[PREM];
  __shared__ float lb[PREM];
  __shared__ unsigned kp[PREM];
  __shared__ float red[1024];

  int b = blockIdx.x;
  int tid = threadIdx.x, nth = blockDim.x;
  const float* boxes = boxesAll + (size_t)b * NBOX * 4;
  const float* ms    = mscoreAll + (size_t)b * NBOX;
  unsigned long long* keys = keysAll + (size_t)b * NSORT;

  for (int i = tid; i < NSORT; i += nth) {
    if (i < NBOX) {
      keys[i] = ((unsigned long long)ford(ms[i]) << 32) |
                (unsigned)(0xFFFFFFFFu - (unsigned)i);
    } else keys[i] = 0ull;
  }
  bitonic_sort_desc(keys, NSORT, tid, nth);

  // gather top PREM into LDS
  for (int i = tid; i < PREM; i += nth) {
    unsigned long long kkey = keys[i];
    unsigned idx = 0xFFFFFFFFu - (unsigned)(kkey & 0xFFFFFFFFu);
    float s = funord((unsigned)(kkey >> 32));
    sc[i] = s;
    lb[i] = (float)(idx % NCLS);
    const float* bo = boxes + (size_t)idx * 4;
    bx[i * 4 + 0] = bo[0]; bx[i * 4 + 1] = bo[1];
    bx[i * 4 + 2] = bo[2]; bx[i * 4 + 3] = bo[3];
    kp[i] = (s > -0.5f) ? 1u : 0u;
  }
  __syncthreads();

  // max over all selected coords (for per-class offset)
  float lm = -1e30f;
  for (int i = tid; i < PREM * 4; i += nth) lm = fmaxf(lm, bx[i]);
  red[tid] = lm;
  __syncthreads();
  for (int s = nth >> 1; s > 0; s >>= 1) {
    if (tid < s) red[tid] = fmaxf(red[tid], red[tid + s]);
    __syncthreads();
  }
  float offmul = red[0] + 1.0f;
  __syncthreads();

  // greedy NMS in score order
  for (int i = 0; i < PREM; ++i) {
    __syncthreads();
    if (!kp[i]) continue;
    float oi = lb[i] * offmul;
    float ax1 = bx[i * 4 + 0] + oi, ay1 = bx[i * 4 + 1] + oi;
    float ax2 = bx[i * 4 + 2] + oi, ay2 = bx[i * 4 + 3] + oi;
    float areaA = (ax2 - ax1) * (ay2 - ay1);
    for (int j = i + 1 + tid; j < PREM; j += nth) {
      if (!kp[j]) continue;
      float oj = lb[j] * offmul;
      float bx1 = bx[j * 4 + 0] + oj, by1 = bx[j * 4 + 1] + oj;
      float bx2 = bx[j * 4 + 2] + oj, by2 = bx[j * 4 + 3] + oj;
      float areaB = (bx2 - bx1) * (by2 - by1);
      float ix1 = fmaxf(ax1, bx1), iy1 = fmaxf(ay1, by1);
      float ix2 = fminf(ax2, bx2), iy2 = fminf(ay2, by2);
      float ww = fmaxf(ix2 - ix1, 0.f), hh = fmaxf(iy2 - iy1, 0.f);
      float inter = ww * hh;
      float iou = inter / (areaA + areaB - inter + 1e-12f);
      if (iou > NMS_TH) kp[j] = 0u;
    }
  }
  __syncthreads();

  // final top-100 of kept scores (ties -> lower index)
  for (int i = tid; i < PREM; i += nth) {
    float s = kp[i] ? sc[i] : -1.0f;
    keys[i] = ((unsigned long long)ford(s) << 32) |
              (unsigned)(0xFFFFFFFFu - (unsigned)i);
  }
  bitonic_sort_desc(keys, PREM, tid, nth);

  float oh = (float)orig_sizes[b * 2 + 0], ow = (float)orig_sizes[b * 2 + 1];
  float hr = oh * (RSCALE / (float)FH);
  float wr = ow * (RSCALE / (float)FW);
  for (int k = tid; k < POSTK; k += nth) {
    unsigned long long kkey = keys[k];
    unsigned idx = 0xFFFFFFFFu - (unsigned)(kkey & 0xFFFFFFFFu);
    float s = funord((unsigned)(kkey >> 32));
    float* o = out + ((size_t)b * POSTK + k) * 6;
    if (s > 0.0f) {
      o[0] = bx[idx * 4 + 0] * wr;  o[1] = bx[idx * 4 + 1] * hr;
      o[2] = bx[idx * 4 + 2] * wr;  o[3] = bx[idx * 4 + 3] * hr;
      o[4] = s;                     o[5] = lb[idx];
    } else {
      o[0] = 0.f; o[1] = 0.f; o[2] = 0.f; o[3] = 0.f; o[4] = 0.f; o[5] = 0.f;
    }
  }
}

// --------------------------------------------------------------------- launch
extern "C" void kernel_launch(void* const* d_in, const int* in_sizes, int n_in,
                              void* d_out, int out_size, void* d_ws, size_t ws_size,
                              hipStream_t stream) {
  const float* feat   = (const float*)d_in[0];
  const float* rois   = (const float*)d_in[1];
  const float* W1     = (const float*)d_in[2];
  const float* b1     = (const float*)d_in[3];
  const float* W2     = (const float*)d_in[4];
  const float* b2     = (const float*)d_in[5];
  const float* Wloc   = (const float*)d_in[6];
  const float* bloc   = (const float*)d_in[7];
  const float* Wcls   = (const float*)d_in[8];
  const float* bcls   = (const float*)d_in[9];
  const int* img_sz   = (const int*)d_in[10];
  const int* orig_sz  = (const int*)d_in[11];
  float* out = (float*)d_out;

  float* ws = (float*)d_ws;
  size_t off = 0;
  float* X      = ws + off;  off += (size_t)BNR * FIN;       // 12,845,056
  float* H1     = ws + off;  off += (size_t)BNR * HID;       //  1,048,576
  float* H2     = ws + off;  off += (size_t)BNR * HID;       //  1,048,576
  float* locs   = ws + off;  off += (size_t)BNR * NCLS * 4;  //     86,016
  float* probs  = ws + off;  off += (size_t)BNR * NCLS;      //     21,504
  float* boxes  = ws + off;  off += (size_t)BATCH * NBOX * 4;//     86,016
  float* mscore = ws + off;  off += (size_t)BATCH * NBOX;    //     21,504
  unsigned long long* keys = (unsigned long long*)(ws + off); // 2*16384 u64

  roi_pool_kernel<<<BNR, 256, 0, stream>>>(feat, rois, X);

  gemm_bias_relu<<<dim3(HID / BNT, BNR / BMT), 256, 0, stream>>>(
      X, W1, b1, H1, BNR, HID, FIN, 1);
  gemm_bias_relu<<<dim3(HID / BNT, BNR / BMT), 256, 0, stream>>>(
      H1, W2, b2, H2, BNR, HID, HID, 1);

  heads_kernel<<<BNR, 128, 0, stream>>>(H2, Wloc, bloc, Wcls, bcls, locs, probs);

  decode_kernel<<<(BATCH * NBOX + 255) / 256, 256, 0, stream>>>(
      rois, locs, probs, img_sz, boxes, mscore);

  nms_kernel<<<BATCH, 1024, 0, stream>>>(boxes, mscore, orig_sz, keys, out);
}